// AdaptiveGeoHypergraph_29712583754327
// MI455X (gfx1250) — compile-verified
//
#include <hip/hip_runtime.h>
#include <math.h>

// ---- problem constants (match reference) ----
#define B_   4
#define N_   10000
#define K1_  17
#define P_   3
#define F_   128
#define H_   128
#define SH_  32
#define BN   (B_ * N_)
#define LN_EPS    1e-5f
#define DEG_CLAMP 1e-6f

typedef float v2f __attribute__((ext_vector_type(2)));
typedef float v8f __attribute__((ext_vector_type(8)));

// ---------------------------------------------------------------------------
// Generic fp32 WMMA GEMM: C[M,Ncols] = rowscale(A[M,K]) @ B[K,Ncols] + bias
// One wave32 computes one 16x16 D tile via V_WMMA_F32_16X16X4_F32.
// A frag (16x4 f32, 2 VGPR): lane<16 holds row l, K=k0/k0+1 ; lane>=16 K=k0+2/+3
// B frag (4x16 f32, 2 VGPR): VGPR r, half hi -> row K=k0+2*hi+r, col = lane&15
// C/D (16x16 f32, 8 VGPR):   VGPR r, half hi -> row r+8*hi,     col = lane&15
// rowscale (1/deg) applied to A rows => scales the matching D rows.
// ---------------------------------------------------------------------------
__global__ void k_gemm_wmma(const float* __restrict__ A, int lda,
                            const float* __restrict__ Bm, int ldb,
                            float* __restrict__ C, int ldc,
                            int M, int Ncols, int K,
                            const float* __restrict__ bias,
                            const float* __restrict__ rowscale) {
  const int ntiles = Ncols >> 4;
  const int tiles  = (M >> 4) * ntiles;
  const int wave   = blockIdx.x * (blockDim.x >> 5) + (threadIdx.x >> 5);
  if (wave >= tiles) return;                      // wave-uniform: EXEC all-1 past here
  const int mtile = wave / ntiles;
  const int ntile = wave % ntiles;
  const int lane  = threadIdx.x & 31;             // wave32
  const int hi    = lane >> 4;
  const int l15   = lane & 15;
  const int rowA  = (mtile << 4) + l15;           // A row this lane feeds
  const int col   = (ntile << 4) + l15;           // D column this lane owns
  const float rs  = rowscale ? rowscale[rowA] : 1.0f;

  v8f acc;
  const float bv = bias ? bias[col] : 0.0f;
#pragma unroll
  for (int r = 0; r < 8; ++r) acc[r] = bv;

  const float* Ap = A + (size_t)rowA * lda;
#pragma unroll 4
  for (int k0 = 0; k0 < K; k0 += 4) {
    const int ka = k0 + (hi << 1);
    v2f a, b;
    a[0] = Ap[ka + 0] * rs;
    a[1] = Ap[ka + 1] * rs;
    b[0] = Bm[(size_t)(ka + 0) * ldb + col];
    b[1] = Bm[(size_t)(ka + 1) * ldb + col];
    // (neg_a, A, neg_b, B, c_mod, C, reuse_a, reuse_b)
    acc = __builtin_amdgcn_wmma_f32_16x16x4_f32(false, a, false, b,
                                                (short)0, acc, false, false);
  }

#pragma unroll
  for (int r = 0; r < 8; ++r)
    C[(size_t)((mtile << 4) + r + (hi << 3)) * ldc + col] = acc[r];
}

// ---------------------------------------------------------------------------
__global__ void k_zero(float* __restrict__ p, int n) {
  int i = blockIdx.x * blockDim.x + threadIdx.x;
  if (i < n) p[i] = 0.0f;
}

// Pack W1 (262x32) into W1s (128x64) and W1p (3x64):
//   cols 0..31  -> u_c path: rows 6..133 (s_c) / rows 0..2 (p_c)
//   cols 32..63 -> u_n path: rows 134..261 (s_n) / rows 3..5 (p_n)
__global__ void k_pack_w1(const float* __restrict__ W1,
                          float* __restrict__ W1s, float* __restrict__ W1p) {
  int i = blockIdx.x * blockDim.x + threadIdx.x;
  if (i < F_ * 64) {
    int k = i >> 6, j = i & 63;
    W1s[i] = (j < SH_) ? W1[(6 + k) * SH_ + j]
                       : W1[(6 + H_ + k) * SH_ + (j - SH_)];
  } else if (i < F_ * 64 + P_ * 64) {
    int t = i - F_ * 64;
    int r = t >> 6, j = t & 63;
    W1p[t] = (j < SH_) ? W1[r * SH_ + j]
                       : W1[(P_ + r) * SH_ + (j - SH_)];
  }
}

// u[b,n,0:32] += p[n]@W1p_c + b1 ; u[b,n,32:64] += p[n]@W1p_n
__global__ void k_u_fix(float* __restrict__ u, const float* __restrict__ pos,
                        const float* __restrict__ W1p, const float* __restrict__ b1) {
  int i = blockIdx.x * blockDim.x + threadIdx.x;
  if (i >= BN * 64) return;
  int j   = i & 63;
  int row = i >> 6;
  int n   = row % N_;
  float v = (j < SH_) ? b1[j] : 0.0f;
#pragma unroll
  for (int t = 0; t < P_; ++t) v += pos[n * P_ + t] * W1p[t * 64 + j];
  u[i] += v;
}

// One wave per node: lane k<17 scores neighbor k, wave-softmax, deg scatter.
__global__ void k_scores(const float* __restrict__ u, const int* __restrict__ nbr,
                         const float* __restrict__ W2, const float* __restrict__ b2,
                         float* __restrict__ alpha, float* __restrict__ deg) {
  int wave = blockIdx.x * (blockDim.x >> 5) + (threadIdx.x >> 5);
  if (wave >= BN) return;
  int lane = threadIdx.x & 31;
  int b = wave / N_, n = wave % N_;
  float score = -3.4e38f;
  int nb = 0;
  if (lane < K1_) {
    nb = nbr[n * K1_ + lane];
    const float* uc = u + (size_t)wave * 64;
    const float* un = u + ((size_t)b * N_ + nb) * 64 + SH_;
    float acc = b2[0];
#pragma unroll
    for (int j = 0; j < SH_; ++j) acc += fmaxf(uc[j] + un[j], 0.0f) * W2[j];
    score = acc;
  }
  float m = score;
  for (int off = 16; off; off >>= 1) m = fmaxf(m, __shfl_xor(m, off, 32));
  float e = (lane < K1_) ? __expf(score - m) : 0.0f;
  float s = e;
  for (int off = 16; off; off >>= 1) s += __shfl_xor(s, off, 32);
  float al = e / s;
  if (lane < K1_) {
    alpha[(size_t)wave * K1_ + lane] = al;
    atomicAdd(&deg[b * N_ + nb], al);
  }
}

__global__ void k_clamp_inv(const float* __restrict__ deg, float* __restrict__ invdeg) {
  int i = blockIdx.x * blockDim.x + threadIdx.x;
  if (i < BN) invdeg[i] = 1.0f / fmaxf(deg[i], DEG_CLAMP);
}

// One 128-thread block per node: edge = sum_k alpha_k * h[nbr_k]; then
// z[nbr_k] += alpha_k * edge (matches reference scatter-add semantics).
__global__ void k_edge_scatter(const float* __restrict__ alpha, const int* __restrict__ nbr,
                               const float* __restrict__ h, float* __restrict__ z) {
  int node = blockIdx.x;
  int d    = threadIdx.x;                 // 0..127
  int b = node / N_, n = node % N_;
  __shared__ float s_a[K1_];
  __shared__ int   s_i[K1_];
  if (d < K1_) {
    s_a[d] = alpha[(size_t)node * K1_ + d];
    s_i[d] = nbr[n * K1_ + d];
  }
  __syncthreads();
  float ed = 0.0f;
#pragma unroll
  for (int k = 0; k < K1_; ++k)
    ed += s_a[k] * h[((size_t)b * N_ + s_i[k]) * H_ + d];
#pragma unroll
  for (int k = 0; k < K1_; ++k)
    atomicAdd(&z[((size_t)b * N_ + s_i[k]) * H_ + d], s_a[k] * ed);
}

// One wave per row: LayerNorm (E[x^2]-mean^2, matches jnp.var) + affine + relu.
__global__ void k_ln_relu(const float* __restrict__ y, const float* __restrict__ g,
                          const float* __restrict__ be, float* __restrict__ out) {
  int row = blockIdx.x * (blockDim.x >> 5) + (threadIdx.x >> 5);
  if (row >= BN) return;
  int lane = threadIdx.x & 31;
  const float* yr = y + (size_t)row * H_;
  float v[4], sum = 0.0f, sq = 0.0f;
#pragma unroll
  for (int t = 0; t < 4; ++t) {
    v[t] = yr[lane + 32 * t];
    sum += v[t];
    sq  += v[t] * v[t];
  }
  for (int off = 16; off; off >>= 1) {
    sum += __shfl_xor(sum, off, 32);
    sq  += __shfl_xor(sq,  off, 32);
  }
  float mean = sum * (1.0f / H_);
  float var  = sq  * (1.0f / H_) - mean * mean;
  float rstd = rsqrtf(var + LN_EPS);
  float* o = out + (size_t)row * H_;
#pragma unroll
  for (int t = 0; t < 4; ++t) {
    int d = lane + 32 * t;
    o[d] = fmaxf((v[t] - mean) * rstd * g[d] + be[d], 0.0f);
  }
}

// ---------------------------------------------------------------------------
extern "C" void kernel_launch(void* const* d_in, const int* in_sizes, int n_in,
                              void* d_out, int out_size, void* d_ws, size_t ws_size,
                              hipStream_t stream) {
  const float* x   = (const float*)d_in[0];   // (B,N,F)
  const float* pos = (const float*)d_in[1];   // (N,P)
  const int*   nbr = (const int*)  d_in[2];   // (N,K1)
  const float* Ws  = (const float*)d_in[3];   // (F,H)
  const float* bs  = (const float*)d_in[4];   // (H)
  const float* W1  = (const float*)d_in[5];   // (262,32)
  const float* b1  = (const float*)d_in[6];   // (32)
  const float* W2  = (const float*)d_in[7];   // (32,1)
  const float* b2  = (const float*)d_in[8];   // (1)
  const float* Wc0 = (const float*)d_in[9];
  const float* bc0 = (const float*)d_in[10];
  const float* Wc1 = (const float*)d_in[11];
  const float* bc1 = (const float*)d_in[12];
  const float* g0  = (const float*)d_in[13];
  const float* be0 = (const float*)d_in[14];
  const float* g1  = (const float*)d_in[15];
  const float* be1 = (const float*)d_in[16];
  float* out = (float*)d_out;

  // Workspace layout (floats). ~95 MB total; resident in MI455X 192 MB L2.
  float* ws = (float*)d_ws;
  size_t o = 0;
  float* s      = ws + o; o += (size_t)BN * H_;       // 5.12M
  float* u      = ws + o; o += (size_t)BN * 64;       // 2.56M
  float* alpha  = ws + o; o += (size_t)BN * K1_;      // 0.68M
  float* deg    = ws + o; o += BN;
  float* invdeg = ws + o; o += BN;
  float* W1s    = ws + o; o += (size_t)F_ * 64;
  float* W1p    = ws + o; o += (size_t)P_ * 64;
  float* z      = ws + o; o += (size_t)BN * H_;       // 5.12M
  float* y      = ws + o; o += (size_t)BN * H_;       // 5.12M
  float* h1     = ws + o; o += (size_t)BN * H_;       // 5.12M
  (void)ws_size; (void)n_in; (void)in_sizes; (void)out_size;

  const int TB = 256;
  auto gemm_blocks = [](int M, int Ncols) {
    int tiles = (M >> 4) * (Ncols >> 4);
    return (tiles + 3) / 4;                    // 4 waves (128 thr) per block
  };

  // 0) clear deg
  k_zero<<<(BN + TB - 1) / TB, TB, 0, stream>>>(deg, BN);
  // 1) pack W1 -> W1s (128x64), W1p (3x64)
  k_pack_w1<<<(F_ * 64 + P_ * 64 + TB - 1) / TB, TB, 0, stream>>>(W1, W1s, W1p);
  // 2) s = x @ Ws + bs            (WMMA, M=40000 K=128 N=128)
  k_gemm_wmma<<<gemm_blocks(BN, H_), 128, 0, stream>>>(
      x, F_, Ws, H_, s, H_, BN, H_, F_, bs, nullptr);
  // 3) u = s @ W1s                (WMMA, M=40000 K=128 N=64)
  k_gemm_wmma<<<gemm_blocks(BN, 64), 128, 0, stream>>>(
      s, H_, W1s, 64, u, 64, BN, 64, H_, nullptr, nullptr);
  // 4) u += positional part + b1
  k_u_fix<<<(BN * 64 + TB - 1) / TB, TB, 0, stream>>>(u, pos, W1p, b1);
  // 5) scores -> softmax alpha, scatter deg
  k_scores<<<(BN + 7) / 8, TB, 0, stream>>>(u, nbr, W2, b2, alpha, deg);
  // 6) invdeg = 1 / max(deg, 1e-6)
  k_clamp_inv<<<(BN + TB - 1) / TB, TB, 0, stream>>>(deg, invdeg);

  // ---- layer 1: h = x ----
  k_zero<<<(BN * H_ + TB - 1) / TB, TB, 0, stream>>>(z, BN * H_);
  k_edge_scatter<<<BN, H_, 0, stream>>>(alpha, nbr, x, z);
  k_gemm_wmma<<<gemm_blocks(BN, H_), 128, 0, stream>>>(
      z, H_, Wc0, H_, y, H_, BN, H_, H_, bc0, invdeg);   // (z/deg) @ Wc0 + bc0
  k_ln_relu<<<(BN + 7) / 8, TB, 0, stream>>>(y, g0, be0, h1);

  // ---- layer 2: h = h1 ----
  k_zero<<<(BN * H_ + TB - 1) / TB, TB, 0, stream>>>(z, BN * H_);
  k_edge_scatter<<<BN, H_, 0, stream>>>(alpha, nbr, h1, z);
  k_gemm_wmma<<<gemm_blocks(BN, H_), 128, 0, stream>>>(
      z, H_, Wc1, H_, y, H_, BN, H_, H_, bc1, invdeg);   // (z/deg) @ Wc1 + bc1
  k_ln_relu<<<(BN + 7) / 8, TB, 0, stream>>>(y, g1, be1, out);
}